// NegSimHead_41918880809474
// MI455X (gfx1250) — compile-verified
//
#include <hip/hip_runtime.h>
#include <math.h>

#define NROWS 8192
#define NC    256

typedef __attribute__((ext_vector_type(2))) float v2f;
typedef __attribute__((ext_vector_type(4))) float v4f;
typedef __attribute__((ext_vector_type(8))) float v8f;

// ---------------------------------------------------------------------------
// Init output: scalars (accumulated via atomics) -> 0, recall arrays -> const,
// mean_rk arrays -> 0.  Layout (flat, return order):
//  [0] neg_sim, [1..4] stds, then 8 arrays of 8192:
//  v_r1, v_r5, v_r10, v_mrk, t_r1, t_r5, t_r10, t_mrk
// ---------------------------------------------------------------------------
__global__ void init_out_kernel(float* __restrict__ out, int out_size) {
    int idx = blockIdx.x * blockDim.x + threadIdx.x;
    if (idx >= out_size) return;
    float val = 0.0f;
    if (idx >= 5) {
        int seg = ((idx - 5) / NROWS) & 3;   // 0:r1 1:r5 2:r10 3:mrk
        val = (seg == 0) ? 1.0f : (seg == 1) ? 5.0f : (seg == 2) ? 10.0f : 0.0f;
    }
    out[idx] = val;
}

// ---------------------------------------------------------------------------
// Row L2-normalize: one wave32 per row (8 elems/lane), wave shuffle reduction.
// ---------------------------------------------------------------------------
__global__ void normalize_kernel(const float* __restrict__ x, float* __restrict__ y) {
    int row  = (blockIdx.x * blockDim.x + threadIdx.x) >> 5;
    int lane = threadIdx.x & 31;
    if (row >= NROWS) return;
    const float* rp = x + (size_t)row * NC;
    float v[8];
    float ss = 0.0f;
#pragma unroll
    for (int q = 0; q < 8; ++q) { v[q] = rp[lane + 32 * q]; ss += v[q] * v[q]; }
#pragma unroll
    for (int off = 16; off > 0; off >>= 1) ss += __shfl_xor(ss, off, 32);
    float inv = 1.0f / fmaxf(sqrtf(ss), 1e-12f);
    float* op = y + (size_t)row * NC;
#pragma unroll
    for (int q = 0; q < 8; ++q) op[lane + 32 * q] = v[q] * inv;
}

// ---------------------------------------------------------------------------
// mean over columns of column-std (ddof=1).  One block per column.
// ---------------------------------------------------------------------------
__global__ void colstd_kernel(const float* __restrict__ x, float* __restrict__ out_scalar) {
    int col = blockIdx.x;
    int tid = threadIdx.x;
    float s = 0.0f, s2 = 0.0f;
    for (int r = tid; r < NROWS; r += 256) {
        float v = x[(size_t)r * NC + col];
        s += v; s2 += v * v;
    }
    __shared__ float sh[256], sh2[256];
    sh[tid] = s; sh2[tid] = s2;
    __syncthreads();
    for (int off = 128; off > 0; off >>= 1) {
        if (tid < off) { sh[tid] += sh[tid + off]; sh2[tid] += sh2[tid + off]; }
        __syncthreads();
    }
    if (tid == 0) {
        float n   = (float)NROWS;
        float var = fmaxf((sh2[0] - sh[0] * sh[0] / n) / (n - 1.0f), 0.0f);
        atomicAdd(out_scalar, sqrtf(var) * (1.0f / NC));
    }
}

// ---------------------------------------------------------------------------
// neg_sim_loss = -0.5/N * ( sum(pv*t) + sum(pt*v) )   (on normalized inputs)
// ---------------------------------------------------------------------------
__global__ void negsim_kernel(const float* __restrict__ vn, const float* __restrict__ tn,
                              const float* __restrict__ pvn, const float* __restrict__ ptn,
                              float* __restrict__ out0) {
    int idx = blockIdx.x * blockDim.x + threadIdx.x;
    int stride = gridDim.x * blockDim.x;
    float s = 0.0f;
    const int total = NROWS * NC;
    for (int i = idx; i < total; i += stride)
        s += pvn[i] * tn[i] + ptn[i] * vn[i];
    __shared__ float sh[256];
    sh[threadIdx.x] = s;
    __syncthreads();
    for (int off = 128; off > 0; off >>= 1) {
        if (threadIdx.x < off) sh[threadIdx.x] += sh[threadIdx.x + off];
        __syncthreads();
    }
    if (threadIdx.x == 0) atomicAdd(out0, sh[0] * (-0.5f / NROWS));
}

// ---------------------------------------------------------------------------
// Diagonals: diag1[i] = <vn_i, ptn_i>,  diag2[i] = <tn_i, pvn_i>
// ---------------------------------------------------------------------------
__global__ void diag_kernel(const float* __restrict__ vn, const float* __restrict__ tn,
                            const float* __restrict__ pvn, const float* __restrict__ ptn,
                            float* __restrict__ diag1, float* __restrict__ diag2) {
    int row  = (blockIdx.x * blockDim.x + threadIdx.x) >> 5;
    int lane = threadIdx.x & 31;
    if (row >= NROWS) return;
    size_t base = (size_t)row * NC;
    float s1 = 0.0f, s2 = 0.0f;
#pragma unroll
    for (int q = 0; q < 8; ++q) {
        int c = lane + 32 * q;
        s1 += vn[base + c] * ptn[base + c];
        s2 += tn[base + c] * pvn[base + c];
    }
#pragma unroll
    for (int off = 16; off > 0; off >>= 1) {
        s1 += __shfl_xor(s1, off, 32);
        s2 += __shfl_xor(s2, off, 32);
    }
    if (lane == 0) { diag1[row] = s1; diag2[row] = s2; }
}

// ---------------------------------------------------------------------------
// Fused WMMA fp32 GEMM (S = A * B^T, A,B [8192x256] row-major) + per-row
// count of S[i,j] > diag[i] (j != i).  mean_rk[i] += count (T==1).
//
// Block = 256 threads (8 wave32), output tile 256x128.  Each wave owns a
// 64x64 region = 4x4 tiles of 16x16 (16 x v8f accumulators, 128 VGPRs).
//
// K trick: within one V_WMMA_F32_16X16X4_F32, lane-half 0 supplies K-slots
// {0,1} and lane-half 1 supplies K-slots {2,3} for all 16 rows.  Since GEMM
// is K-order invariant, we let half0 carry logical K = k0..k0+3 and half1
// carry k0+4..k0+7 (same permutation for A and B -> exact result).  One
// global_load_b128 per fragment then feeds TWO WMMAs: per K-step of 8 a wave
// issues 8 b128 loads and 32 WMMAs (128 B of L2 traffic per 2048-FLOP WMMA,
// 4 WMMAs per VMEM instruction).
// ---------------------------------------------------------------------------
__global__ void __launch_bounds__(256)
wmma_gemm_count_kernel(const float* __restrict__ A, const float* __restrict__ B,
                       const float* __restrict__ diag, float* __restrict__ mrk) {
    const int tid  = threadIdx.x;
    const int wave = tid >> 5;        // 0..7
    const int lane = tid & 31;
    const int m    = lane & 15;       // row/col within fragment
    const int h4   = (lane >> 4) * 4; // logical-K base offset for this half

    const int rowB = blockIdx.y * 256;
    const int colB = blockIdx.x * 128;
    const int wr   = wave >> 1;       // 0..3 -> 64-row band
    const int wc   = wave & 1;        // 0..1 -> 64-col band
    const int waveRow = rowB + wr * 64;
    const int waveCol = colB + wc * 64;

    const v8f zero = {0.f, 0.f, 0.f, 0.f, 0.f, 0.f, 0.f, 0.f};
    v8f acc[4][4];
#pragma unroll
    for (int i = 0; i < 4; ++i)
#pragma unroll
        for (int j = 0; j < 4; ++j) acc[i][j] = zero;

    const float* aPtr[4];
    const float* bPtr[4];
#pragma unroll
    for (int rt = 0; rt < 4; ++rt)
        aPtr[rt] = A + (size_t)(waveRow + rt * 16 + m) * NC + h4;
#pragma unroll
    for (int ct = 0; ct < 4; ++ct)
        bPtr[ct] = B + (size_t)(waveCol + ct * 16 + m) * NC + h4;

    for (int k0 = 0; k0 < NC; k0 += 8) {
        v4f af[4], bf[4];
#pragma unroll
        for (int rt = 0; rt < 4; ++rt) af[rt] = *(const v4f*)(aPtr[rt] + k0);
#pragma unroll
        for (int ct = 0; ct < 4; ++ct) bf[ct] = *(const v4f*)(bPtr[ct] + k0);

#pragma unroll
        for (int rt = 0; rt < 4; ++rt) {
            v2f a_lo = __builtin_shufflevector(af[rt], af[rt], 0, 1);
            v2f a_hi = __builtin_shufflevector(af[rt], af[rt], 2, 3);
#pragma unroll
            for (int ct = 0; ct < 4; ++ct) {
                v2f b_lo = __builtin_shufflevector(bf[ct], bf[ct], 0, 1);
                v2f b_hi = __builtin_shufflevector(bf[ct], bf[ct], 2, 3);
                acc[rt][ct] = __builtin_amdgcn_wmma_f32_16x16x4_f32(
                    false, a_lo, false, b_lo, (short)0, acc[rt][ct], false, false);
                acc[rt][ct] = __builtin_amdgcn_wmma_f32_16x16x4_f32(
                    false, a_hi, false, b_hi, (short)0, acc[rt][ct], false, false);
            }
        }
    }

    // ---- count S[i,j] > diag[i], j != i ----
    __shared__ int cnt[256];
    cnt[tid] = 0;
    __syncthreads();

    const int hi8 = (lane >> 4) * 8;  // C/D layout: VGPR r holds M=r (lanes 0-15), M=r+8 (lanes 16-31)
    float dv[4][8];
#pragma unroll
    for (int rt = 0; rt < 4; ++rt)
#pragma unroll
        for (int r = 0; r < 8; ++r)
            dv[rt][r] = diag[waveRow + rt * 16 + r + hi8];

#pragma unroll
    for (int rt = 0; rt < 4; ++rt) {
#pragma unroll
        for (int ct = 0; ct < 4; ++ct) {
            int j = waveCol + ct * 16 + m;
#pragma unroll
            for (int r = 0; r < 8; ++r) {
                int i = waveRow + rt * 16 + r + hi8;
                bool pred = (acc[rt][ct][r] > dv[rt][r]) && (j != i);
                unsigned bm = (unsigned)__ballot(pred);
                if (lane == 0) {
                    int lr = wr * 64 + rt * 16 + r;
                    atomicAdd(&cnt[lr],     __popc(bm & 0xFFFFu)); // rows M=r
                    atomicAdd(&cnt[lr + 8], __popc(bm >> 16));     // rows M=r+8
                }
            }
        }
    }
    __syncthreads();
    atomicAdd(&mrk[rowB + tid], (float)cnt[tid]);
}

// ---------------------------------------------------------------------------
extern "C" void kernel_launch(void* const* d_in, const int* in_sizes, int n_in,
                              void* d_out, int out_size, void* d_ws, size_t ws_size,
                              hipStream_t stream) {
    const float* v_feat = (const float*)d_in[0];
    const float* t_feat = (const float*)d_in[1];
    const float* p_v    = (const float*)d_in[2];
    const float* p_t    = (const float*)d_in[3];
    float* out = (float*)d_out;

    // workspace: 4 normalized matrices (32 MB) + 2 diagonals (64 KB)
    const size_t MAT = (size_t)NROWS * NC;
    float* ws    = (float*)d_ws;
    float* vn    = ws;
    float* tn    = ws + MAT;
    float* pvn   = ws + 2 * MAT;
    float* ptn   = ws + 3 * MAT;
    float* diag1 = ws + 4 * MAT;
    float* diag2 = diag1 + NROWS;

    init_out_kernel<<<(out_size + 255) / 256, 256, 0, stream>>>(out, out_size);

    normalize_kernel<<<NROWS / 8, 256, 0, stream>>>(v_feat, vn);
    normalize_kernel<<<NROWS / 8, 256, 0, stream>>>(t_feat, tn);
    normalize_kernel<<<NROWS / 8, 256, 0, stream>>>(p_v,    pvn);
    normalize_kernel<<<NROWS / 8, 256, 0, stream>>>(p_t,    ptn);

    colstd_kernel<<<NC, 256, 0, stream>>>(vn,  out + 1);
    colstd_kernel<<<NC, 256, 0, stream>>>(tn,  out + 2);
    colstd_kernel<<<NC, 256, 0, stream>>>(pvn, out + 3);
    colstd_kernel<<<NC, 256, 0, stream>>>(ptn, out + 4);

    negsim_kernel<<<512, 256, 0, stream>>>(vn, tn, pvn, ptn, out);

    diag_kernel<<<NROWS / 8, 256, 0, stream>>>(vn, tn, pvn, ptn, diag1, diag2);

    dim3 grid(NROWS / 128, NROWS / 256);
    // s1 = v @ pt^T  -> v_mrk at offset 5 + 3*N
    wmma_gemm_count_kernel<<<grid, 256, 0, stream>>>(vn, ptn, diag1, out + 5 + 3 * NROWS);
    // s2 = t @ pv^T  -> t_mrk at offset 5 + 7*N
    wmma_gemm_count_kernel<<<grid, 256, 0, stream>>>(tn, pvn, diag2, out + 5 + 7 * NROWS);
}